// Instance_bantch_64622077935626
// MI455X (gfx1250) — compile-verified
//
#include <hip/hip_runtime.h>

// ---------------------------------------------------------------------------
// SimCLR NT-Xent loss, fused (never materializes the 16384x16384 sim matrix).
// f32 WMMA (v_wmma_f32_16x16x4_f32) for the Gram matrix + online logsumexp.
// Round 2: 4 independent WMMA accumulators (break C RAW chain) and
//          single-exp online-LSE updates (halve v_exp_f32 pressure).
// ---------------------------------------------------------------------------

typedef float v2f __attribute__((ext_vector_type(2)));
typedef float v8f __attribute__((ext_vector_type(8)));

#define BATCH   4096
#define NROW    (2 * BATCH)      // 8192  ("N" in the reference)
#define MROW    (4 * BATCH)      // 16384 rows of z = concat(z_i, z_j)
#define DIM     128
#define INV_T   2.0f             // 1 / temperature (T = 0.5)

#define PANEL    32              // columns of z staged per LDS panel
#define LSTRIDE  132             // padded dword stride per column (DIM + 4)
#define NPANELS  (MROW / PANEL)  // 512
#define WGSIZE   128             // 4 waves (wave32)
#define ROWS_PER_WAVE 16
#define ROWS_PER_WG   64
#define NEG_SENTINEL  (-1.0e30f) // finite "-inf" (avoids inf-inf NaN in LSE)

// --------------------------- positives ------------------------------------
// pos[i] = pos[i+N] = dot(z_i[i], z_j[i]) / T  for i in [0, N)
__global__ __launch_bounds__(256) void pos_kernel(const float* __restrict__ zi,
                                                  const float* __restrict__ zj,
                                                  float* __restrict__ pos) {
  const int i = blockIdx.x * blockDim.x + threadIdx.x;
  if (i >= NROW) return;
  const float4* a = (const float4*)(zi + (size_t)i * DIM);
  const float4* b = (const float4*)(zj + (size_t)i * DIM);
  float acc = 0.0f;
#pragma unroll
  for (int k = 0; k < DIM / 4; ++k) {
    const float4 x = a[k], y = b[k];
    acc += x.x * y.x + x.y * y.y + x.z * y.z + x.w * y.w;
  }
  acc *= INV_T;
  pos[i] = acc;
  pos[i + NROW] = acc;
}

// ----------------------- fused Gram + masked row LSE ----------------------
__global__ __launch_bounds__(WGSIZE) void simclr_neg_lse_kernel(
    const float* __restrict__ zi, const float* __restrict__ zj,
    float* __restrict__ neg_lse) {
  __shared__ float sB[2][PANEL * LSTRIDE];

  const int tid  = threadIdx.x;
  const int lane = tid & 31;
  const int wave = tid >> 5;
  const int lo16 = lane & 15;
  const int hi   = lane >> 4;  // 0 or 1 (which 16-lane half)

  const int r0 = blockIdx.x * ROWS_PER_WG + wave * ROWS_PER_WAVE;

  // ---- preload A fragments: rows [r0, r0+16), K = 0..127 ----
  // A(16x4 f32) layout: lane holds (M = lane%16, K = 2*(lane/16) + {0,1})
  const float* abase = (r0 < NROW) ? (zi + (size_t)r0 * DIM)
                                   : (zj + (size_t)(r0 - NROW) * DIM);
  const float* aptr = abase + lo16 * DIM + 2 * hi;
  v2f a[32];
#pragma unroll
  for (int ks = 0; ks < 32; ++ks) a[ks] = *(const v2f*)(aptr + ks * 4);

  // ---- per-lane online logsumexp state (8 rows per lane) ----
  float m[8], s[8];
#pragma unroll
  for (int r = 0; r < 8; ++r) { m[r] = NEG_SENTINEL; s[r] = 0.0f; }

  float4 stage[8];
  // prologue: fetch panel 0 (columns 0..31 live in z_i) and park it in LDS
  {
#pragma unroll
    for (int u = 0; u < 8; ++u)
      stage[u] = *(const float4*)(zi + u * 512 + tid * 4);
#pragma unroll
    for (int u = 0; u < 8; ++u) {
      const int l = u * 512 + tid * 4;
      *(float4*)&sB[0][(l >> 7) * LSTRIDE + (l & 127)] = stage[u];
    }
  }

  for (int p = 0; p < NPANELS; ++p) {
    const int buf = p & 1;
    __syncthreads();  // panel p visible in sB[buf]; sB[buf^1] free to fill

    // prefetch next panel into registers (latency overlaps the WMMA work)
    if (p + 1 < NPANELS) {
      const int c0n = (p + 1) * PANEL;
      const float* pb = (c0n < NROW) ? (zi + (size_t)c0n * DIM)
                                     : (zj + (size_t)(c0n - NROW) * DIM);
#pragma unroll
      for (int u = 0; u < 8; ++u)
        stage[u] = *(const float4*)(pb + u * 512 + tid * 4);
    }

    const int c0 = p * PANEL;
#pragma unroll
    for (int ct = 0; ct < 2; ++ct) {
      // B(4x16) layout: lane holds (K = 2*(lane/16) + {0,1}, N = lane%16)
      const int pc = ct * 16 + lo16;  // column within panel
      const float* bptr = &sB[buf][pc * LSTRIDE + 2 * hi];

      // 4 independent accumulators -> WMMA dependency distance of 4
      v8f acc0 = {}, acc1 = {}, acc2 = {}, acc3 = {};
#pragma unroll
      for (int ks = 0; ks < 32; ks += 4) {
        const v2f b0 = *(const v2f*)(bptr + (ks + 0) * 4);
        const v2f b1 = *(const v2f*)(bptr + (ks + 1) * 4);
        const v2f b2 = *(const v2f*)(bptr + (ks + 2) * 4);
        const v2f b3 = *(const v2f*)(bptr + (ks + 3) * 4);
        acc0 = __builtin_amdgcn_wmma_f32_16x16x4_f32(false, a[ks + 0], false,
                                                     b0, (short)0, acc0, false, false);
        acc1 = __builtin_amdgcn_wmma_f32_16x16x4_f32(false, a[ks + 1], false,
                                                     b1, (short)0, acc1, false, false);
        acc2 = __builtin_amdgcn_wmma_f32_16x16x4_f32(false, a[ks + 2], false,
                                                     b2, (short)0, acc2, false, false);
        acc3 = __builtin_amdgcn_wmma_f32_16x16x4_f32(false, a[ks + 3], false,
                                                     b3, (short)0, acc3, false, false);
      }

      // C layout: VGPR r -> row (r + 8*hi), lanes' N = lane%16
      const int j = c0 + ct * 16 + lo16;  // global column for this lane
#pragma unroll
      for (int r = 0; r < 8; ++r) {
        const int i = r0 + r + 8 * hi;    // global row
        float v = ((acc0[r] + acc1[r]) + (acc2[r] + acc3[r])) * INV_T;
        const int d = i - j;
        if (d == 0 || d == NROW || d == -NROW) v = NEG_SENTINEL;  // mask
        // single-exp online LSE: one of the two exponents is always exp(0)=1
        const float dlt  = m[r] - v;
        const float e    = __expf(fminf(dlt, -dlt));  // exp(-|dlt|)
        const bool  keep = (dlt >= 0.0f);
        s[r] = keep ? (s[r] + e) : (s[r] * e + 1.0f);
        m[r] = keep ? m[r] : v;
      }
    }

    // commit prefetched panel into the other LDS buffer
    if (p + 1 < NPANELS) {
#pragma unroll
      for (int u = 0; u < 8; ++u) {
        const int l = u * 512 + tid * 4;
        *(float4*)&sB[buf ^ 1][(l >> 7) * LSTRIDE + (l & 127)] = stage[u];
      }
    }
  }

  // ---- merge the 16 lanes that share each row (xor within 16-lane halves) ----
#pragma unroll
  for (int off = 1; off < 16; off <<= 1) {
#pragma unroll
    for (int r = 0; r < 8; ++r) {
      const float mo = __shfl_xor(m[r], off, 32);
      const float so = __shfl_xor(s[r], off, 32);
      const float dlt  = m[r] - mo;
      const float e    = __expf(fminf(dlt, -dlt));  // exp(-|dlt|)
      const bool  keep = (dlt >= 0.0f);
      s[r] = keep ? (s[r] + so * e) : (s[r] * e + so);
      m[r] = keep ? m[r] : mo;
    }
  }
  if (lo16 == 0) {  // lane 0 -> rows r0..r0+7, lane 16 -> rows r0+8..r0+15
#pragma unroll
    for (int r = 0; r < 8; ++r)
      neg_lse[r0 + r + 8 * hi] = m[r] + __logf(s[r]);
  }
}

// --------------------------- final scalar reduce --------------------------
__global__ __launch_bounds__(256) void loss_kernel(const float* __restrict__ pos,
                                                   const float* __restrict__ nlse,
                                                   float* __restrict__ out) {
  __shared__ float red[256];
  float acc = 0.0f;
  for (int i = threadIdx.x; i < MROW; i += 256) {
    const float p = pos[i], q = nlse[i];
    const float dlt = p - q;
    const float e   = __expf(fminf(dlt, -dlt));     // exp(-|p-q|)
    const float lse = fmaxf(p, q) + log1pf(e);      // logaddexp(p, q)
    acc += lse - p;
  }
  red[threadIdx.x] = acc;
  __syncthreads();
  for (int st = 128; st > 0; st >>= 1) {
    if (threadIdx.x < st) red[threadIdx.x] += red[threadIdx.x + st];
    __syncthreads();
  }
  if (threadIdx.x == 0) out[0] = red[0] / (float)NROW;
}

// ---------------------------------------------------------------------------
extern "C" void kernel_launch(void* const* d_in, const int* in_sizes, int n_in,
                              void* d_out, int out_size, void* d_ws, size_t ws_size,
                              hipStream_t stream) {
  (void)in_sizes; (void)n_in; (void)out_size; (void)ws_size;
  const float* zi = (const float*)d_in[0];  // [2B, 128] f32
  const float* zj = (const float*)d_in[1];  // [2B, 128] f32
  float* pos = (float*)d_ws;                // MROW floats
  float* nls = pos + MROW;                  // MROW floats
  float* out = (float*)d_out;               // scalar f32 loss

  pos_kernel<<<NROW / 256, 256, 0, stream>>>(zi, zj, pos);
  simclr_neg_lse_kernel<<<MROW / ROWS_PER_WG, WGSIZE, 0, stream>>>(zi, zj, nls);
  loss_kernel<<<1, 256, 0, stream>>>(pos, nls, out);
}